// SinkhornDivergence_56573309223924
// MI455X (gfx1250) — compile-verified
//
#include <hip/hip_runtime.h>
#include <hip/hip_bf16.h>

// ---------------------------------------------------------------------------
// Sinkhorn divergence on MI455X (gfx1250).
//   K = exp(-sqdist(x0,x1)/eps)  built once with f32 WMMA (v_wmma_f32_16x16x4_f32)
//   100 Sinkhorn iterations run in ONE persistent kernel (grid barrier); K
//   (64MB) is L2-resident (192MB L2), so all 200 GEMVs stream from L2.
//   The dense vector (u or v) is staged in LDS each half-iteration so WGP$
//   capacity is devoted to streaming K.
//   Final cost recomputes M via WMMA and does a deterministic 2-stage reduce.
// ---------------------------------------------------------------------------

#define NN 4096
#define DD 64

static constexpr float kEpsInv  = 1.0f / 0.1f;     // 1/EPSILON
static constexpr float kEpsDiv  = 1e-8f;
static constexpr int   kIters   = 100;
static constexpr float kAVal    = 1.0f / (float)NN;
static constexpr float kBVal    = 1.0f / (float)NN;

static constexpr int kTilesPerDim = NN / 16;                      // 256
static constexpr int kNumTiles    = kTilesPerDim * kTilesPerDim;  // 65536
static constexpr int kGemmWavesPerBlk = 8;                        // 256 thr
static constexpr int kGemmBlocks  = kNumTiles / kGemmWavesPerBlk; // 8192

static constexpr int kIterBlocks  = 128;
static constexpr int kIterThreads = 512;                 // 16 waves
static constexpr int kRowsPerBlk  = NN / kIterBlocks;    // 32
static constexpr int kRowsPerWave = kRowsPerBlk / 16;    // 2
static constexpr int kColsPerBlk  = NN / kIterBlocks;    // 32
static constexpr int kGroups      = kIterThreads / kColsPerBlk; // 16

// Workspace layout (in floats)
static constexpr size_t K_OFF    = 0;
static constexpr size_t SQ0_OFF  = (size_t)NN * NN;      // 16M
static constexpr size_t SQ1_OFF  = SQ0_OFF + NN;
static constexpr size_t U_OFF    = SQ1_OFF + NN;
static constexpr size_t V_OFF    = U_OFF + NN;
static constexpr size_t PART_OFF = V_OFF + NN;
static constexpr size_t CNT_OFF  = PART_OFF + kGemmBlocks;

typedef __attribute__((ext_vector_type(2))) float v2f;
typedef __attribute__((ext_vector_type(8))) float v8f;

// ---------------------------------------------------------------------------
// Init: row squared-norms of x0/x1 (one wave per row), u=v=1, barrier cnt=0.
// ---------------------------------------------------------------------------
__global__ void sink_init(const float* __restrict__ x0,
                          const float* __restrict__ x1,
                          float* __restrict__ ws) {
  const int gtid = blockIdx.x * blockDim.x + threadIdx.x;
  const int wid  = gtid >> 5;            // global wave id, 0..8191
  const int lane = threadIdx.x & 31;

  const float* src = (wid < NN) ? x0 : x1;
  const int row    = (wid < NN) ? wid : wid - NN;
  const float a0 = src[row * DD + lane];
  const float a1 = src[row * DD + lane + 32];
  float s = a0 * a0 + a1 * a1;
  #pragma unroll
  for (int off = 16; off > 0; off >>= 1) s += __shfl_xor(s, off, 32);
  if (lane == 0) ws[((wid < NN) ? SQ0_OFF : SQ1_OFF) + row] = s;

  if (gtid < NN)            ws[U_OFF + gtid] = 1.0f;
  else if (gtid < 2 * NN)   ws[V_OFF + gtid - NN] = 1.0f;
  if (gtid == 0)            ((unsigned*)(ws + CNT_OFF))[0] = 0u;
}

// ---------------------------------------------------------------------------
// Build K with f32 WMMA: each wave owns one 16x16 tile of x0 @ x1^T,
// reduced over K=64 in sixteen 16x16x4 steps, then K=exp(-max(M,0)/eps).
// ---------------------------------------------------------------------------
__global__ void __launch_bounds__(256)
sink_buildK(const float* __restrict__ x0,
            const float* __restrict__ x1,
            float* __restrict__ ws) {
  const int wave  = (blockIdx.x * blockDim.x + threadIdx.x) >> 5;
  const int lane  = threadIdx.x & 31;
  const int tileM = wave >> 8;           // /256
  const int tileN = wave & 255;
  const int half  = lane >> 4;           // 0 or 1
  const int l15   = lane & 15;

  // A: 16x4 f32, lane = row (l15), half selects K-pair {0,1} vs {2,3}
  const float* arow = x0 + (size_t)(tileM * 16 + l15) * DD + 2 * half;
  // B: 4x16 f32 = x1^T tile, lane = col, same K striping
  const float* brow = x1 + (size_t)(tileN * 16 + l15) * DD + 2 * half;

  v8f c = {};
  #pragma unroll
  for (int kk = 0; kk < DD; kk += 4) {
    v2f a = *(const v2f*)(arow + kk);
    v2f b = *(const v2f*)(brow + kk);
    c = __builtin_amdgcn_wmma_f32_16x16x4_f32(false, a, false, b,
                                              (short)0, c, false, false);
  }

  const float sq1v = ws[SQ1_OFF + tileN * 16 + l15];  // per-lane column norm
  #pragma unroll
  for (int r = 0; r < 8; ++r) {
    const int row = tileM * 16 + r + 8 * half;        // D-matrix VGPR layout
    float m = ws[SQ0_OFF + row] + sq1v - 2.0f * c[r];
    m = fmaxf(m, 0.0f);
    ws[K_OFF + (size_t)row * NN + tileN * 16 + l15] = __expf(-m * kEpsInv);
  }
}

// ---------------------------------------------------------------------------
// Grid-wide barrier: monotonic arrival counter, no reset needed.
// ---------------------------------------------------------------------------
__device__ inline void grid_sync(unsigned* cnt) {
  __syncthreads();
  if (threadIdx.x == 0) {
    __threadfence();
    const unsigned arrive = atomicAdd(cnt, 1u) + 1u;
    const unsigned target =
        ((arrive + (unsigned)kIterBlocks - 1u) / (unsigned)kIterBlocks) *
        (unsigned)kIterBlocks;
    while (__hip_atomic_load(cnt, __ATOMIC_ACQUIRE,
                             __HIP_MEMORY_SCOPE_AGENT) < target) {
      __builtin_amdgcn_s_sleep(1);
    }
    __threadfence();
  }
  __syncthreads();
}

// ---------------------------------------------------------------------------
// Persistent Sinkhorn iteration kernel: 128 blocks x 512 threads.
//   Each half-iteration stages the dense vector (v or u) into LDS (16KB).
//   u-update: one wave per row (x2), float4 lane-strided K loads (L2-hot),
//             vector read from LDS.
//   v-update: lane = column (32 consecutive cols -> 128B coalesced per wave),
//             16-way row split + fixed-order LDS tree reduce.
// ---------------------------------------------------------------------------
__global__ void __launch_bounds__(kIterThreads, 1)
sink_iterate(float* __restrict__ ws) {
  const float* __restrict__ Kp = ws + K_OFF;
  float* __restrict__ u = ws + U_OFF;
  float* __restrict__ v = ws + V_OFF;
  unsigned* cnt = (unsigned*)(ws + CNT_OFF);

  const int tid  = threadIdx.x;
  const int wv   = tid >> 5;                      // 0..15
  const int lane = tid & 31;
  const int col  = blockIdx.x * kColsPerBlk + (tid & (kColsPerBlk - 1));
  const int grp  = tid / kColsPerBlk;             // 0..15

  __shared__ float vec[NN];                       // 16KB staged vector
  __shared__ float red[kIterThreads];

  for (int it = 0; it < kIters; ++it) {
    // ---- stage v into LDS ----
    for (int i = tid; i < NN; i += kIterThreads) vec[i] = v[i];
    __syncthreads();

    // ---- u = a / (K v + eps) : rows [blockIdx.x*32, +32), wave per row ----
    #pragma unroll
    for (int rr = 0; rr < kRowsPerWave; ++rr) {
      const int row = blockIdx.x * kRowsPerBlk + wv * kRowsPerWave + rr;
      const float4* __restrict__ kr = (const float4*)(Kp + (size_t)row * NN);
      const float4* __restrict__ v4 = (const float4*)vec;
      float acc = 0.0f;
      #pragma unroll 8
      for (int j = lane; j < NN / 4; j += 32) {
        const float4 kk = kr[j];
        const float4 vv = v4[j];
        acc += kk.x * vv.x + kk.y * vv.y + kk.z * vv.z + kk.w * vv.w;
      }
      #pragma unroll
      for (int off = 16; off > 0; off >>= 1) acc += __shfl_xor(acc, off, 32);
      if (lane == 0) u[row] = kAVal / (acc + kEpsDiv);
    }
    grid_sync(cnt);

    // ---- stage u into LDS ----
    for (int i = tid; i < NN; i += kIterThreads) vec[i] = u[i];
    __syncthreads();

    // ---- v = b / (K^T u + eps) : cols [blockIdx.x*32, +32) ----
    {
      float acc = 0.0f;
      #pragma unroll 8
      for (int i = grp; i < NN; i += kGroups)
        acc += Kp[(size_t)i * NN + col] * vec[i];
      red[tid] = acc;
      __syncthreads();
      #pragma unroll
      for (int s = kIterThreads / 2; s >= kColsPerBlk; s >>= 1) {
        if (tid < s) red[tid] += red[tid + s];
        __syncthreads();
      }
      if (tid < kColsPerBlk) v[col] = kBVal / (red[tid] + kEpsDiv);
    }
    if (it != kIters - 1) grid_sync(cnt);
  }
}

// ---------------------------------------------------------------------------
// cost = sum_ij u_i * K_ij * v_j * M_ij ; M recomputed per tile via WMMA.
// Per-block fixed-order LDS reduce -> partials[blockIdx.x].
// ---------------------------------------------------------------------------
__global__ void __launch_bounds__(256)
sink_cost(const float* __restrict__ x0,
          const float* __restrict__ x1,
          float* __restrict__ ws) {
  const int tid   = threadIdx.x;
  const int wave  = (blockIdx.x * blockDim.x + tid) >> 5;
  const int lane  = tid & 31;
  const int tileM = wave >> 8;
  const int tileN = wave & 255;
  const int half  = lane >> 4;
  const int l15   = lane & 15;

  const float* arow = x0 + (size_t)(tileM * 16 + l15) * DD + 2 * half;
  const float* brow = x1 + (size_t)(tileN * 16 + l15) * DD + 2 * half;

  v8f c = {};
  #pragma unroll
  for (int kk = 0; kk < DD; kk += 4) {
    v2f a = *(const v2f*)(arow + kk);
    v2f b = *(const v2f*)(brow + kk);
    c = __builtin_amdgcn_wmma_f32_16x16x4_f32(false, a, false, b,
                                              (short)0, c, false, false);
  }

  const int colg   = tileN * 16 + l15;
  const float sq1v = ws[SQ1_OFF + colg];
  const float vj   = ws[V_OFF + colg];
  float part = 0.0f;
  #pragma unroll
  for (int r = 0; r < 8; ++r) {
    const int row = tileM * 16 + r + 8 * half;
    float m = ws[SQ0_OFF + row] + sq1v - 2.0f * c[r];
    m = fmaxf(m, 0.0f);
    const float k = __expf(-m * kEpsInv);
    part += ws[U_OFF + row] * k * vj * m;
  }

  __shared__ float red[256];
  red[tid] = part;
  __syncthreads();
  #pragma unroll
  for (int s = 128; s > 0; s >>= 1) {
    if (tid < s) red[tid] += red[tid + s];
    __syncthreads();
  }
  if (tid == 0) ws[PART_OFF + blockIdx.x] = red[0];
}

__global__ void sink_reduce(const float* __restrict__ ws,
                            float* __restrict__ out) {
  __shared__ float red[256];
  float s = 0.0f;
  for (int i = threadIdx.x; i < kGemmBlocks; i += 256)
    s += ws[PART_OFF + i];
  red[threadIdx.x] = s;
  __syncthreads();
  #pragma unroll
  for (int k = 128; k > 0; k >>= 1) {
    if (threadIdx.x < k) red[threadIdx.x] += red[threadIdx.x + k];
    __syncthreads();
  }
  if (threadIdx.x == 0) out[0] = red[0];
}

// ---------------------------------------------------------------------------
extern "C" void kernel_launch(void* const* d_in, const int* in_sizes, int n_in,
                              void* d_out, int out_size, void* d_ws,
                              size_t ws_size, hipStream_t stream) {
  const float* x0 = (const float*)d_in[0];
  const float* x1 = (const float*)d_in[1];
  float* ws  = (float*)d_ws;
  float* out = (float*)d_out;

  sink_init<<<(2 * NN) / 8, 256, 0, stream>>>(x0, x1, ws);          // 1024 blks
  sink_buildK<<<kGemmBlocks, 256, 0, stream>>>(x0, x1, ws);         // 8192 blks
  sink_iterate<<<kIterBlocks, kIterThreads, 0, stream>>>(ws);       // persistent
  sink_cost<<<kGemmBlocks, 256, 0, stream>>>(x0, x1, ws);
  sink_reduce<<<1, 256, 0, stream>>>(ws, out);
}